// AgreementRouting_37211596652835
// MI455X (gfx1250) — compile-verified
//
#include <hip/hip_runtime.h>

// Problem constants (from reference setup_inputs)
#define B_N    64
#define N_L    2048
#define N_H    32
#define DD     16
#define CHUNKS 16                 // partial-s rows (matches WMMA K=16 chain)
#define WPB    8                  // waves per block
#define LPW    16                 // l values per wave: CHUNKS*WPB*LPW == N_L
#define NCOL   (B_N * N_H * DD)   // 32768 columns of the partial matrix

typedef float v2f __attribute__((ext_vector_type(2)));
typedef float v8f __attribute__((ext_vector_type(8)));

__device__ __forceinline__ float dot16(const float* a, const float* b) {
  float s = 0.f;
#pragma unroll
  for (int i = 0; i < 16; ++i) s = fmaf(a[i], b[i], s);
  return s;
}

// One fused routing pass: for every (b,l) compute logits = b[l,h] + sum_j <u, v_j>,
// softmax over h (32 lanes), accumulate c*u into per-wave partial s, block-reduce,
// write one partial row P[chunk, b*512 + h*16 + d]. Reads u exactly once.
template <int K>
__global__ __launch_bounds__(256) void routing_pass(
    const float* __restrict__ u, const float* __restrict__ blog,
    const float* __restrict__ v0, const float* __restrict__ v1,
    const float* __restrict__ v2, float* __restrict__ P)
{
  const int lane  = threadIdx.x & 31;
  const int wave  = threadIdx.x >> 5;
  const int chunk = blockIdx.x;   // 0..CHUNKS-1
  const int bb    = blockIdx.y;   // 0..B_N-1
  const int h     = lane;         // lane == output capsule

  // Cache v_j[b,h,:] per lane (tiny, register resident)
  float vr0[16], vr1[16], vr2[16];
  if (K > 0) {
    const float4* p = (const float4*)(v0 + ((size_t)bb * N_H + h) * DD);
#pragma unroll
    for (int q = 0; q < 4; ++q) { float4 t = p[q];
      vr0[4*q]=t.x; vr0[4*q+1]=t.y; vr0[4*q+2]=t.z; vr0[4*q+3]=t.w; }
  }
  if (K > 1) {
    const float4* p = (const float4*)(v1 + ((size_t)bb * N_H + h) * DD);
#pragma unroll
    for (int q = 0; q < 4; ++q) { float4 t = p[q];
      vr1[4*q]=t.x; vr1[4*q+1]=t.y; vr1[4*q+2]=t.z; vr1[4*q+3]=t.w; }
  }
  if (K > 2) {
    const float4* p = (const float4*)(v2 + ((size_t)bb * N_H + h) * DD);
#pragma unroll
    for (int q = 0; q < 4; ++q) { float4 t = p[q];
      vr2[4*q]=t.x; vr2[4*q+1]=t.y; vr2[4*q+2]=t.z; vr2[4*q+3]=t.w; }
  }

  float acc[16];
#pragma unroll
  for (int q = 0; q < 16; ++q) acc[q] = 0.f;

  const int l0 = chunk * (WPB * LPW) + wave * LPW;
  const float* rowp = u + (((size_t)bb * N_L + l0) * N_H + h) * DD;

  for (int i = 0; i < LPW; ++i) {
    const int l = l0 + i;
    __builtin_prefetch(rowp + 2 * N_H * DD, 0, 1);  // gfx1250 global_prefetch_b8
    float ur[16];
    const float4* p4 = (const float4*)rowp;          // 64B/lane, coalesced b128
#pragma unroll
    for (int q = 0; q < 4; ++q) { float4 t = p4[q];
      ur[4*q]=t.x; ur[4*q+1]=t.y; ur[4*q+2]=t.z; ur[4*q+3]=t.w; }

    float logit = blog[(size_t)l * N_H + h];
    if (K > 0) logit += dot16(ur, vr0);
    if (K > 1) logit += dot16(ur, vr1);
    if (K > 2) logit += dot16(ur, vr2);

    // softmax over the 32 output capsules == the 32 lanes of this wave
    float m = logit;
#pragma unroll
    for (int msk = 16; msk >= 1; msk >>= 1) m = fmaxf(m, __shfl_xor(m, msk, 32));
    const float e = __expf(logit - m);
    float ssum = e;
#pragma unroll
    for (int msk = 16; msk >= 1; msk >>= 1) ssum += __shfl_xor(ssum, msk, 32);
    const float c = e / ssum;

#pragma unroll
    for (int q = 0; q < 16; ++q) acc[q] = fmaf(c, ur[q], acc[q]);
    rowp += N_H * DD;
  }

  // deterministic block reduction of 8 wave-partials
  __shared__ float red[WPB * N_H * DD];   // 16 KB
#pragma unroll
  for (int q = 0; q < 16; ++q) red[wave * (N_H * DD) + h * DD + q] = acc[q];
  __syncthreads();
  for (int e = threadIdx.x; e < N_H * DD; e += 256) {
    float s = 0.f;
#pragma unroll
    for (int w = 0; w < WPB; ++w) s += red[w * (N_H * DD) + e];
    P[(size_t)chunk * NCOL + (size_t)bb * (N_H * DD) + e] = s;
  }
}

// Cross-chunk reduction as a real matmul: s[:,col] = ones[1x16] x P[16 x NCOL],
// done with chained V_WMMA_F32_16X16X4_F32 (A = 1.0). One wave per capsule:
// the chain's 16 columns are exactly that capsule's d=0..15, so the same wave
// finishes squash_hinton with a width-16 shuffle reduction.
__global__ __launch_bounds__(256) void reduce_squash(
    const float* __restrict__ P, float* __restrict__ out)
{
  const int lane = threadIdx.x & 31;
  const int wave = threadIdx.x >> 5;
  const int cap  = blockIdx.x * WPB + wave;     // 0..B_N*N_H-1
  const int col  = cap * 16 + (lane & 15);      // column = (b,h,d) flat index
  const int rb   = (lane >= 16) ? 2 : 0;        // lane-half K offset in B layout

  v2f a; a[0] = 1.0f; a[1] = 1.0f;              // ones row => column sums
  v8f c = {};
#pragma unroll
  for (int o = 0; o < 16; o += 4) {
    v2f bv;
    bv[0] = P[(size_t)(o + rb    ) * NCOL + col];
    bv[1] = P[(size_t)(o + rb + 1) * NCOL + col];
    c = __builtin_amdgcn_wmma_f32_16x16x4_f32(
        false, a, false, bv, (short)0, c, false, false);
  }
  const float s = c[0];                          // D[m,n] identical over m

  // squash: ||s||^2 over d=16 (reduce within half-wave)
  float sq = s * s;
#pragma unroll
  for (int msk = 8; msk >= 1; msk >>= 1) sq += __shfl_xor(sq, msk, 16);
  const float v = s * (sq / (1.0f + sq)) * rsqrtf(sq + 1e-8f);
  if (lane < 16) out[col] = v;
}

extern "C" void kernel_launch(void* const* d_in, const int* in_sizes, int n_in,
                              void* d_out, int out_size, void* d_ws, size_t ws_size,
                              hipStream_t stream)
{
  const float* u    = (const float*)d_in[0];
  const float* blog = (const float*)d_in[1];

  float* v0 = (float*)d_ws;            // [B,NH,D]
  float* v1 = v0 + NCOL;
  float* v2 = v1 + NCOL;
  float* P  = v2 + NCOL;               // [CHUNKS, NCOL] partial s

  dim3 gA(CHUNKS, B_N), bA(256);
  dim3 gB((B_N * N_H) / WPB), bB(256);

  routing_pass<0><<<gA, bA, 0, stream>>>(u, blog, v0, v1, v2, P);
  reduce_squash<<<gB, bB, 0, stream>>>(P, v0);
  routing_pass<1><<<gA, bA, 0, stream>>>(u, blog, v0, v1, v2, P);
  reduce_squash<<<gB, bB, 0, stream>>>(P, v1);
  routing_pass<2><<<gA, bA, 0, stream>>>(u, blog, v0, v1, v2, P);
  reduce_squash<<<gB, bB, 0, stream>>>(P, v2);
  routing_pass<3><<<gA, bA, 0, stream>>>(u, blog, v0, v1, v2, P);
  reduce_squash<<<gB, bB, 0, stream>>>(P, (float*)d_out);
}